// QPUnrolledNetwork_27135603376520
// MI455X (gfx1250) — compile-verified
//
#include <hip/hip_runtime.h>
#include <hip/hip_bf16.h>

// ---- problem constants (from reference) ----
#define INPUT_SIZE 512
#define HIDDEN     1024
#define N_QP       64
#define M_QP       128
#define QP_ITER    50
#define N_P        2080          // 64*65/2
#define N_H        8192          // 128*64
#define N_OUT      10464         // N_P + N_QP + N_H + M_QP
#define BATCH      1024

typedef __attribute__((ext_vector_type(2))) float v2f;
typedef __attribute__((ext_vector_type(4))) float v4f;
typedef __attribute__((ext_vector_type(8))) float v8f;
typedef __attribute__((ext_vector_type(4))) int   i4;

#if defined(__gfx1250__) && __has_builtin(__builtin_amdgcn_global_load_async_to_lds_b128)
#define HAVE_ASYNC_LDS 1
#else
#define HAVE_ASYNC_LDS 0
#endif

#if HAVE_ASYNC_LDS
typedef __attribute__((address_space(3))) i4 lds_i4;

static __device__ __forceinline__ void wait_async0() {
#if __has_builtin(__builtin_amdgcn_s_wait_asynccnt)
    __builtin_amdgcn_s_wait_asynccnt(0);
#else
    asm volatile("s_wait_asynccnt 0" ::: "memory");
#endif
}
#endif

// ============================================================
// GEMM: C = act(A @ B + bias), A: MxK row-major, B: KxN row-major.
// 128 threads = 4 waves; block tile 64x128; wave tile 32x64 (2x4 WMMA).
// f32 16x16x4 WMMA; K chunked by 16 through double-buffered LDS.
// A tile: async global->LDS b128 copies (contiguous in K, ASYNCcnt).
// B tile: register-staged, stored K-pair-interleaved so WMMA fragments
//         are single b64 LDS loads.
// ============================================================
__global__ __launch_bounds__(128) void gemm_bias_act(
    const float* __restrict__ A, const float* __restrict__ B,
    const float* __restrict__ bias, float* __restrict__ C,
    int M, int N, int K, int do_relu)
{
    constexpr int TM = 64, TN = 128, TK = 16, AP = 20; // AP: padded A stride
    __shared__ __align__(16) float As[2][TM][AP];
    __shared__ __align__(16) float Bs[2][TK / 2][TN][2];

    const int t     = threadIdx.x;
    const int wave  = t >> 5;
    const int lane  = t & 31;
    const int m16   = lane & 15;            // M (A) / N (B,C) index within 16
    const int khalf = (lane >> 4) << 1;     // 0 for lanes 0-15, 2 for lanes 16-31

    const int wrow = (wave & 1) * 32;       // wave's row offset in block tile
    const int wcol = (wave >> 1) * 64;      // wave's col offset in block tile

    const int m0 = blockIdx.y * TM;
    const int n0 = blockIdx.x * TN;

    // ---- staging / copy index helpers ----
    const int ar  = t >> 2;                 // 0..31 (rows ar, ar+32)  [fallback A path]
    const int ac  = (t & 3) * 4;            // 0..12
    const int rp  = t >> 5;                 // 0..3  (k-pairs rp, rp+4)
    const int c4  = (t & 31) * 4;           // 0..124

#if HAVE_ASYNC_LDS
    auto asyncLoadA = [&](int k0, int buf) {
        #pragma unroll
        for (int p = 0; p < 2; ++p) {
            const int s   = p * 128 + t;    // segment 0..255
            const int row = s >> 2;         // 0..63
            const int seg = (s & 3) * 4;    // 0,4,8,12
            const float* g = A + (size_t)(m0 + row) * K + k0 + seg;
            __builtin_amdgcn_global_load_async_to_lds_b128(
                (i4*)g,
                (lds_i4*)&As[buf][row][seg],
                0, 0);
        }
    };
#else
    auto loadA = [&](int k0, v4f a[2]) {
        a[0] = *(const v4f*)(A + (size_t)(m0 + ar)      * K + k0 + ac);
        a[1] = *(const v4f*)(A + (size_t)(m0 + ar + 32) * K + k0 + ac);
    };
    auto storeA = [&](int buf, v4f a[2]) {
        *(v4f*)&As[buf][ar][ac]      = a[0];
        *(v4f*)&As[buf][ar + 32][ac] = a[1];
    };
#endif
    auto loadB = [&](int k0, v4f b[2][2]) {
        const int col = n0 + c4;
        #pragma unroll
        for (int p = 0; p < 2; ++p) {
            const int k = k0 + 2 * (rp + p * 4);
            if (col + 3 < N) {
                b[p][0] = *(const v4f*)(B + (size_t)k * N + col);
                b[p][1] = *(const v4f*)(B + (size_t)(k + 1) * N + col);
            } else {
                #pragma unroll
                for (int i = 0; i < 4; ++i) {
                    b[p][0][i] = (col + i < N) ? B[(size_t)k * N + col + i] : 0.0f;
                    b[p][1][i] = (col + i < N) ? B[(size_t)(k + 1) * N + col + i] : 0.0f;
                }
            }
        }
    };
    auto storeB = [&](int buf, v4f b[2][2]) {
        #pragma unroll
        for (int p = 0; p < 2; ++p) {
            const int pi = rp + p * 4;
            #pragma unroll
            for (int i = 0; i < 4; ++i) {
                v2f w;
                w.x = b[p][0][i];
                w.y = b[p][1][i];
                *(v2f*)&Bs[buf][pi][c4 + i][0] = w;
            }
        }
    };

    v8f acc[2][4] = {};
    const int nchunks = K / TK;

    {   // prologue: chunk 0 into buffer 0
#if HAVE_ASYNC_LDS
        asyncLoadA(0, 0);
#else
        v4f a0[2];
        loadA(0, a0);
        storeA(0, a0);
#endif
        v4f b0[2][2];
        loadB(0, b0);
        storeB(0, b0);
#if HAVE_ASYNC_LDS
        wait_async0();                        // own async copies landed
#endif
    }

    for (int c = 0; c < nchunks; ++c) {
        __syncthreads();                      // buf (c&1) ready across waves
        const int buf = c & 1;
        const bool more = (c + 1 < nchunks);

        v4f bN[2][2];
#if HAVE_ASYNC_LDS
        if (more) {                           // next A goes straight to LDS (async)
            asyncLoadA((c + 1) * TK, buf ^ 1);
            loadB((c + 1) * TK, bN);
        }
#else
        v4f aN[2];
        if (more) {
            loadA((c + 1) * TK, aN);
            loadB((c + 1) * TK, bN);
        }
#endif

        #pragma unroll
        for (int kk = 0; kk < TK; kk += 4) {
            const int kb = kk + khalf;        // even
            v2f af[2], bf[4];
            #pragma unroll
            for (int mi = 0; mi < 2; ++mi) {
                const int r = wrow + mi * 16 + m16;
                af[mi] = *(const v2f*)&As[buf][r][kb];
            }
            #pragma unroll
            for (int ni = 0; ni < 4; ++ni) {
                const int cc = wcol + ni * 16 + m16;
                bf[ni] = *(const v2f*)&Bs[buf][kb >> 1][cc][0];
            }
            #pragma unroll
            for (int mi = 0; mi < 2; ++mi)
                #pragma unroll
                for (int ni = 0; ni < 4; ++ni)
                    acc[mi][ni] = __builtin_amdgcn_wmma_f32_16x16x4_f32(
                        false, af[mi], false, bf[ni],
                        (short)0, acc[mi][ni], false, false);
        }

        if (more) {
#if !HAVE_ASYNC_LDS
            storeA(buf ^ 1, aN);
#endif
            storeB(buf ^ 1, bN);
        }
#if HAVE_ASYNC_LDS
        wait_async0();                        // own async A landed before barrier
#endif
    }

    // ---- store: C layout VGPR j -> M = j (lanes 0-15) / j+8 (lanes 16-31), N = lane%16 ----
    #pragma unroll
    for (int mi = 0; mi < 2; ++mi) {
        #pragma unroll
        for (int ni = 0; ni < 4; ++ni) {
            const int col = n0 + wcol + ni * 16 + m16;
            if (col < N) {
                const float bv = bias ? bias[col] : 0.0f;
                #pragma unroll
                for (int j = 0; j < 8; ++j) {
                    const int row = m0 + wrow + mi * 16 + ((lane < 16) ? j : j + 8);
                    float vv = acc[mi][ni][j] + bv;
                    if (do_relu) vv = fmaxf(vv, 0.0f);
                    C[(size_t)row * N + col] = vv;
                }
            }
        }
    }
}

// ============================================================
// make_psd: unpack lower-tri params, softplus diag, P = L L^T
// one workgroup (256 thr) per batch item
// ============================================================
__global__ __launch_bounds__(256) void make_psd_kernel(
    const float* __restrict__ out, float* __restrict__ P)
{
    const int bidx = blockIdx.x;
    const float* p = out + (size_t)bidx * N_OUT;   // P_params at offset 0
    __shared__ float L[64][65];
    const int t = threadIdx.x;

    for (int i = t; i < 64 * 65; i += 256) (&L[0][0])[i] = 0.0f;
    __syncthreads();

    for (int idx = t; idx < N_P; idx += 256) {
        int r = (int)((sqrtf(8.0f * (float)idx + 1.0f) - 1.0f) * 0.5f);
        while ((r + 1) * (r + 2) / 2 <= idx) ++r;
        while (r * (r + 1) / 2 > idx) --r;
        const int c = idx - r * (r + 1) / 2;
        float v = p[idx];
        if (c == r) {
            float sp = (v > 30.0f) ? v : log1pf(expf(v)); // stable softplus
            v = 0.1f + sp;                                // sqrt(0.01) + softplus
        }
        L[r][c] = v;
    }
    __syncthreads();

    float* Pb = P + (size_t)bidx * 64 * 64;
    for (int e = t; e < 4096; e += 256) {
        const int i = e >> 6, j = e & 63;
        const int kmax = (i < j) ? i : j;
        float s = 0.0f;
        for (int k = 0; k <= kmax; ++k) s += L[i][k] * L[j][k];
        Pb[e] = s;
    }
}

// ============================================================
// power iteration on H^T H (10 iters) -> tau = 0.9 / (||H v|| + 1e-6)
// one workgroup (128 thr) per batch item; H resident in LDS (pad 65)
// ============================================================
__global__ __launch_bounds__(128) void power_iter_kernel(
    const float* __restrict__ out, float* __restrict__ tau)
{
    const int bidx = blockIdx.x;
    const int t = threadIdx.x;
    const float* Hg = out + (size_t)bidx * N_OUT + (N_P + N_QP);
    __shared__ float Hs[128][65];
    __shared__ float v[64];
    __shared__ float u[64];
    __shared__ float w[128];
    __shared__ float red[128];

    for (int e = t; e < N_H; e += 128) Hs[e >> 6][e & 63] = Hg[e];
    if (t < 64) v[t] = 0.125f;                 // 1/sqrt(64)
    __syncthreads();

    for (int it = 0; it < 10; ++it) {
        // w = H v (row t)
        float s = 0.0f;
        #pragma unroll 8
        for (int n = 0; n < 64; ++n) s += Hs[t][n] * v[n];
        w[t] = s;
        __syncthreads();
        // u = H^T w (cols 0..63)
        if (t < 64) {
            float s2 = 0.0f;
            #pragma unroll 8
            for (int m = 0; m < 128; ++m) s2 += Hs[m][t] * w[m];
            u[t] = s2;
            red[t] = s2 * s2;
        } else red[t] = 0.0f;
        __syncthreads();
        for (int off = 64; off > 0; off >>= 1) {
            if (t < off) red[t] += red[t + off];
            __syncthreads();
        }
        const float inv = 1.0f / (sqrtf(red[0]) + 1e-12f);
        if (t < 64) v[t] = u[t] * inv;
        __syncthreads();
    }
    // sn = ||H v|| + 1e-6
    float s = 0.0f;
    #pragma unroll 8
    for (int n = 0; n < 64; ++n) s += Hs[t][n] * v[n];
    red[t] = s * s;
    __syncthreads();
    for (int off = 64; off > 0; off >>= 1) {
        if (t < off) red[t] += red[t + off];
        __syncthreads();
    }
    if (t == 0) tau[bidx] = 0.9f / (sqrtf(red[0]) + 1e-6f);
}

// ============================================================
// Minv = inv(I + tau * P), Gauss-Jordan (SPD, no pivoting needed)
// one workgroup (64 thr = columns) per batch item
// ============================================================
__global__ __launch_bounds__(64) void invert_kernel(
    const float* __restrict__ P, const float* __restrict__ tau,
    float* __restrict__ Minv)
{
    const int bidx = blockIdx.x;
    const int t = threadIdx.x;                 // column index
    __shared__ float A[64][65];
    __shared__ float Inv[64][65];
    __shared__ float fcol[64];
    __shared__ float piv;

    const float* Pb = P + (size_t)bidx * 4096;
    const float tv = tau[bidx];
    for (int i = 0; i < 64; ++i) {
        A[i][t]   = ((i == t) ? 1.0f : 0.0f) + tv * Pb[i * 64 + t];
        Inv[i][t] = (i == t) ? 1.0f : 0.0f;
    }
    __syncthreads();

    for (int k = 0; k < 64; ++k) {
        if (t == 0) piv = 1.0f / A[k][k];
        __syncthreads();
        const float fk = A[t][k];              // column-k snapshot
        A[k][t]   *= piv;
        Inv[k][t] *= piv;
        fcol[t] = (t == k) ? 0.0f : fk;
        __syncthreads();
        for (int i = 0; i < 64; ++i) {
            const float f = fcol[i];
            A[i][t]   -= f * A[k][t];
            Inv[i][t] -= f * Inv[k][t];
        }
        __syncthreads();
    }

    float* Mb = Minv + (size_t)bidx * 4096;
    for (int i = 0; i < 64; ++i) Mb[i * 64 + t] = Inv[i][t];
}

// ============================================================
// 50 QP iterations, everything LDS-resident; 128 thr per batch item
// ============================================================
__global__ __launch_bounds__(128) void qp_solve_kernel(
    const float* __restrict__ out, const float* __restrict__ Minv,
    const float* __restrict__ tau, float* __restrict__ xs)
{
    const int bidx = blockIdx.x;
    const int t = threadIdx.x;
    __shared__ float Hs[128][65];
    __shared__ float Mi[64][65];
    __shared__ float xv[64], xb[64], lam[128], qv[64], bv[128], rhs[64];

    const float* ob = out + (size_t)bidx * N_OUT;
    for (int e = t; e < N_H; e += 128)  Hs[e >> 6][e & 63] = ob[N_P + N_QP + e];
    for (int e = t; e < 4096; e += 128) Mi[e >> 6][e & 63] = Minv[(size_t)bidx * 4096 + e];
    if (t < 64) { qv[t] = ob[N_P + t]; xv[t] = 0.0f; xb[t] = 0.0f; }
    bv[t]  = ob[N_P + N_QP + N_H + t];
    lam[t] = 0.0f;
    const float tv = tau[bidx];   // sig == tau
    __syncthreads();

    for (int it = 0; it < QP_ITER; ++it) {
        // lam = relu(lam - sig*(H xbar + b))
        float s = 0.0f;
        #pragma unroll 8
        for (int n = 0; n < 64; ++n) s += Hs[t][n] * xb[n];
        lam[t] = fmaxf(lam[t] - tv * (s + bv[t]), 0.0f);
        __syncthreads();
        // rhs = x + tau*(H^T lam - q)
        if (t < 64) {
            float s2 = 0.0f;
            #pragma unroll 8
            for (int m = 0; m < 128; ++m) s2 += Hs[m][t] * lam[m];
            rhs[t] = xv[t] + tv * (s2 - qv[t]);
        }
        __syncthreads();
        // x = Minv rhs ; xbar = 2x - x_old
        if (t < 64) {
            float s3 = 0.0f;
            #pragma unroll 8
            for (int k = 0; k < 64; ++k) s3 += Mi[t][k] * rhs[k];
            const float xo = xv[t];
            xv[t] = s3;
            xb[t] = 2.0f * s3 - xo;
        }
        __syncthreads();
    }
    if (t < 64) xs[(size_t)bidx * 64 + t] = xv[t];
}

// ============================================================
extern "C" void kernel_launch(void* const* d_in, const int* in_sizes, int n_in,
                              void* d_out, int out_size, void* d_ws, size_t ws_size,
                              hipStream_t stream) {
    const float* x  = (const float*)d_in[0];
    const float* W1 = (const float*)d_in[1];
    const float* b1 = (const float*)d_in[2];
    const float* W2 = (const float*)d_in[3];
    const float* b2 = (const float*)d_in[4];
    const float* W3 = (const float*)d_in[5];
    const float* b3 = (const float*)d_in[6];

    float* ws  = (float*)d_ws;
    float* h1  = ws;                                    // 1024*1024
    float* h2  = h1  + (size_t)BATCH * HIDDEN;          // 1024*1024
    float* out = h2  + (size_t)BATCH * HIDDEN;          // 1024*10464
    float* P   = out + (size_t)BATCH * N_OUT;           // 1024*4096
    float* Mi  = P   + (size_t)BATCH * N_QP * N_QP;     // 1024*4096
    float* tau = Mi  + (size_t)BATCH * N_QP * N_QP;     // 1024

    // MLP: three WMMA GEMMs (block tile 64x128)
    gemm_bias_act<<<dim3(HIDDEN / 128, BATCH / 64), 128, 0, stream>>>(
        x,  W1, b1, h1, BATCH, HIDDEN, INPUT_SIZE, 1);
    gemm_bias_act<<<dim3(HIDDEN / 128, BATCH / 64), 128, 0, stream>>>(
        h1, W2, b2, h2, BATCH, HIDDEN, HIDDEN, 1);
    gemm_bias_act<<<dim3((N_OUT + 127) / 128, BATCH / 64), 128, 0, stream>>>(
        h2, W3, b3, out, BATCH, N_OUT, HIDDEN, 0);

    // per-batch small linear algebra
    make_psd_kernel  <<<BATCH, 256, 0, stream>>>(out, P);
    power_iter_kernel<<<BATCH, 128, 0, stream>>>(out, tau);
    invert_kernel    <<<BATCH,  64, 0, stream>>>(P, tau, Mi);
    qp_solve_kernel  <<<BATCH, 128, 0, stream>>>(out, Mi, tau, (float*)d_out);
}